// GroupFeatureBuilder_90151363543244
// MI455X (gfx1250) — compile-verified
//
#include <hip/hip_runtime.h>
#include <cstdint>

#define N_ROBOTS 2048
#define EMBED    256
#define M_GROUPS 4096
#define GROUP_K  16
#define N_OBS    64
#define OUT_COLS 518   // 256 h_g + 256 h_glob + 1 size + 3 attn + 1 dist + 1 clr

#define HG_BLOCKS 16
#define HG_ROWS   (N_ROBOTS / HG_BLOCKS)

typedef __attribute__((ext_vector_type(2)))  float    v2f;
typedef __attribute__((ext_vector_type(8)))  float    v8f;

__device__ __forceinline__ float wave_sum(float v) {
#pragma unroll
  for (int off = 16; off > 0; off >>= 1) v += __shfl_xor(v, off, 32);
  return v;
}
__device__ __forceinline__ float wave_min(float v) {
#pragma unroll
  for (int off = 16; off > 0; off >>= 1) v = fminf(v, __shfl_xor(v, off, 32));
  return v;
}

// ---- pass 1: rowsum[r] = sum_n attn_rr[r, n]  (collapses the 536MB einsum) ----
__global__ void __launch_bounds__(256) rowsum_kernel(const float* __restrict__ attn_rr,
                                                     float* __restrict__ rowsum) {
  __shared__ float red[256];
  const int r = blockIdx.x;
  const float* row = attn_rr + (size_t)r * N_ROBOTS;
  float s = 0.f;
  for (int c = threadIdx.x; c < N_ROBOTS; c += 256) s += row[c];
  red[threadIdx.x] = s;
  __syncthreads();
  for (int w = 128; w > 0; w >>= 1) {
    if (threadIdx.x < w) red[threadIdx.x] += red[threadIdx.x + w];
    __syncthreads();
  }
  if (threadIdx.x == 0) rowsum[r] = red[0];
}

// ---- pass 2a: per-block partial column sums of h (parallel across 16 WGPs) ----
__global__ void __launch_bounds__(256) hglob_partial_kernel(const float* __restrict__ h,
                                                            float* __restrict__ partial) {
  const int d = threadIdx.x;
  const int r0 = blockIdx.x * HG_ROWS;
  float s = 0.f;
  for (int r = 0; r < HG_ROWS; ++r) s += h[(size_t)(r0 + r) * EMBED + d];
  partial[blockIdx.x * EMBED + d] = s;
}

// ---- pass 2b: deterministic combine (no float atomics -> bit-identical replays) ----
__global__ void __launch_bounds__(256) hglob_combine_kernel(const float* __restrict__ partial,
                                                            float* __restrict__ hglob) {
  const int d = threadIdx.x;
  float s = 0.f;
#pragma unroll
  for (int b = 0; b < HG_BLOCKS; ++b) s += partial[b * EMBED + d];
  hglob[d] = s * (1.0f / (float)N_ROBOTS);
}

// ---- pass 3: one wave32 per group ----
__global__ void __launch_bounds__(256) group_kernel(
    const float* __restrict__ h, const float* __restrict__ attn_rr,
    const float* __restrict__ attn_ro, const float* __restrict__ dist,
    const float* __restrict__ clr, const int* __restrict__ groups,
    const float* __restrict__ rowsum, const float* __restrict__ hglob,
    float* __restrict__ out) {
  const int lane = threadIdx.x & 31;
  const int m = blockIdx.x * 8 + (threadIdx.x >> 5);

  int g[GROUP_K];
  const int* gp = groups + (size_t)m * GROUP_K;
#pragma unroll
  for (int j = 0; j < GROUP_K; ++j) g[j] = gp[j];

  float* orow = out + (size_t)m * OUT_COLS;
  const int n = lane & 15;
  const int half = lane >> 4;

  // ========== h_g = (1/16) * ones(1x16) * H_sub(16x256) via v_wmma_f32_16x16x4_f32 ==========
  // A (16x4 f32) = 1/16 everywhere. Row-uniform A + uniform weights make the result
  // invariant to the exact K <-> (VGPR, lane-half) permutation of the B layout, and
  // every row of D is identical -> both lane-halves hold valid results.
  v2f aw; aw.x = 0.0625f; aw.y = 0.0625f;

  // 8 per-lane gather base pointers (one per B slot), reused with immediate offsets.
  const float* rp[8];
#pragma unroll
  for (int kc = 0; kc < 4; ++kc) {
    rp[2 * kc]     = h + (size_t)g[kc * 4 + 2 * half]     * EMBED + n;
    rp[2 * kc + 1] = h + (size_t)g[kc * 4 + 2 * half + 1] * EMBED + n;
  }

#pragma unroll
  for (int dt = 0; dt < 16; dt += 2) {
    v8f accA = {};
    v8f accB = {};
#pragma unroll
    for (int kc = 0; kc < 4; ++kc) {
      v2f bA, bB;
      bA.x = rp[2 * kc][dt * 16];
      bA.y = rp[2 * kc + 1][dt * 16];
      bB.x = rp[2 * kc][dt * 16 + 16];
      bB.y = rp[2 * kc + 1][dt * 16 + 16];
      accA = __builtin_amdgcn_wmma_f32_16x16x4_f32(false, aw, false, bA,
                                                   (short)0, accA, false, false);
      accB = __builtin_amdgcn_wmma_f32_16x16x4_f32(false, aw, false, bB,
                                                   (short)0, accB, false, false);
    }
    // Full-wave store, no EXEC games: half 0 -> tile dt (M=0 row), half 1 -> tile dt+1
    // (M=8 row, identical values).
    const float val = half ? accB[0] : accA[0];
    orow[(dt + half) * 16 + n] = val;
  }

  // ================= attn stats =================
  // first-occurrence mask over slots (distinct member set), uniform per wave
  unsigned fm = 0;
#pragma unroll
  for (int j = 0; j < 16; ++j) {
    bool f = true;
    for (int i = 0; i < j; ++i) f = f && (g[i] != g[j]);
    fm |= ((unsigned)f) << j;
  }
  const int Dn = __popc(fm);  // # distinct members

  // one pass over the gathered 16x16 sub-matrix feeds BOTH a_in and the member-sum
  float num = 0.f, mem = 0.f, cntf = 0.f;
#pragma unroll
  for (int t = 0; t < 8; ++t) {
    const int p = lane + t * 32;
    const int i = p >> 4, j = p & 15;
    const float s = attn_rr[(size_t)g[i] * N_ROBOTS + g[j]];
    const bool ne = (g[i] != g[j]);
    num  += ne ? s : 0.f;
    cntf += ne ? 1.f : 0.f;
    mem  += ((fm >> j) & 1u) ? s : 0.f;
  }
  num  = wave_sum(num);
  cntf = wave_sum(cntf);
  mem  = wave_sum(mem);
  const float a_in = num / fmaxf(cntf, 1.f);

  float rs = (lane < 16) ? rowsum[g[lane]] : 0.f;
  rs = wave_sum(rs);
  const float n_out = (float)(N_ROBOTS - Dn);
  const float a_out = (rs - mem) / fmaxf((float)GROUP_K * n_out, 1.f);

  float obs = 0.f;
#pragma unroll
  for (int t = 0; t < 32; ++t) {
    const int p = lane + t * 32;
    const int k = p >> 6, c = p & 63;
    obs += attn_ro[(size_t)g[k] * N_OBS + c];
  }
  obs = wave_sum(obs) * (1.f / (float)(GROUP_K * N_OBS));

  float dv = (lane < 16) ? dist[g[lane]] : 0.f;
  dv = wave_sum(dv) * (1.f / (float)GROUP_K);
  float cv = (lane < 16) ? clr[g[lane]] : 3.4e38f;
  cv = wave_min(cv);

  // h_glob broadcast (L2-hot). Row stride 518 floats keeps orow+256 8B-aligned -> float2.
  {
    const float2* hg2 = (const float2*)hglob;
    float2* og2 = (float2*)(orow + EMBED);
#pragma unroll
    for (int t = 0; t < 4; ++t) og2[lane + t * 32] = hg2[lane + t * 32];
  }

  if (lane == 0) {
    orow[512] = (float)GROUP_K / 3.0f;
    orow[513] = a_in;
    orow[514] = a_out;
    orow[515] = obs;
    orow[516] = dv;
    orow[517] = cv;
  }
}

extern "C" void kernel_launch(void* const* d_in, const int* in_sizes, int n_in,
                              void* d_out, int out_size, void* d_ws, size_t ws_size,
                              hipStream_t stream) {
  (void)in_sizes; (void)n_in; (void)out_size; (void)ws_size;
  const float* h       = (const float*)d_in[0];
  const float* attn_rr = (const float*)d_in[1];
  const float* attn_ro = (const float*)d_in[2];
  const float* dist    = (const float*)d_in[3];
  const float* clr     = (const float*)d_in[4];
  const int*   groups  = (const int*)d_in[5];
  float* out = (float*)d_out;

  float* rowsum  = (float*)d_ws;                 // N_ROBOTS floats
  float* partial = rowsum + N_ROBOTS;            // HG_BLOCKS * EMBED floats
  float* hglob   = partial + HG_BLOCKS * EMBED;  // EMBED floats

  rowsum_kernel<<<dim3(N_ROBOTS), dim3(256), 0, stream>>>(attn_rr, rowsum);
  hglob_partial_kernel<<<dim3(HG_BLOCKS), dim3(EMBED), 0, stream>>>(h, partial);
  hglob_combine_kernel<<<dim3(1), dim3(EMBED), 0, stream>>>(partial, hglob);
  group_kernel<<<dim3(M_GROUPS / 8), dim3(256), 0, stream>>>(
      h, attn_rr, attn_ro, dist, clr, groups, rowsum, hglob, out);
}